// GATIRec_46935402611144
// MI455X (gfx1250) — compile-verified
//
#include <hip/hip_runtime.h>
#include <math.h>

// Problem constants (match reference)
#define NN   100000
#define EE   1600000
#define HH   2
#define CC   32
#define HC   64          // H*C
#define LL   4
#define INF_ 4           // input one-hot dim
#define ECLS 5

typedef float v2f __attribute__((ext_vector_type(2)));
typedef float v8f __attribute__((ext_vector_type(8)));

// ---------- helpers ----------
__device__ __forceinline__ unsigned enc_ord(float f) {
  unsigned u = __float_as_uint(f);
  return (u & 0x80000000u) ? ~u : (u | 0x80000000u);
}
__device__ __forceinline__ float dec_ord(unsigned m) {
  return (m & 0x80000000u) ? __uint_as_float(m & 0x7FFFFFFFu) : __uint_as_float(~m);
}
#define ENC_NEG_INF 0x007FFFFFu   // enc_ord(-inf)

// ---------- layer 0 projection: [N,4] @ [4,64] ----------
__global__ void egat_proj0(const float* __restrict__ x, const float* __restrict__ W0,
                           float* __restrict__ hout) {
  int t = blockIdx.x * blockDim.x + threadIdx.x;
  if (t >= NN * HC) return;
  int n = t >> 6, j = t & 63;
  const float* xr = x + n * INF_;
  float s = xr[0] * W0[0 * HC + j] + xr[1] * W0[1 * HC + j]
          + xr[2] * W0[2 * HC + j] + xr[3] * W0[3 * HC + j];
  hout[t] = s;
}

// ---------- layers 1..3 projection: [N,64] @ [64,64] via V_WMMA_F32_16X16X4_F32 ----------
// One wave (32 lanes) computes a 16x64 output tile. N = 100000 = 6250 * 16 exactly.
__global__ __launch_bounds__(32) void egat_proj_wmma(const float* __restrict__ hin,
                                                     const float* __restrict__ W,
                                                     float* __restrict__ hout) {
  const int lane = threadIdx.x & 31;
  const int m  = lane & 15;    // row within tile (A), col within 16-col tile (B/C/D)
  const int kh = lane >> 4;    // lane-half: A k-pair select, C/D row-half select
  const int rowBase = blockIdx.x * 16;

  v8f c0 = {}, c1 = {}, c2 = {}, c3 = {};
  const float* arow = hin + (size_t)(rowBase + m) * HC;

#pragma unroll
  for (int k = 0; k < HC; k += 4) {
    // A fragment 16x4 f32: lane-half kh holds K = k+2*kh (+0,+1) for row m
    v2f a;
    a.x = arow[k + 2 * kh + 0];
    a.y = arow[k + 2 * kh + 1];
    const float* w0 = W + (size_t)(k + 2 * kh + 0) * HC;
    const float* w1 = W + (size_t)(k + 2 * kh + 1) * HC;
    // B fragments 4x16 f32 per 16-col tile: col = nt*16 + m, rows K = k+2*kh (+0,+1)
    v2f b;
    b.x = w0[0 + m];  b.y = w1[0 + m];
    c0 = __builtin_amdgcn_wmma_f32_16x16x4_f32(false, a, false, b, (short)0, c0, false, false);
    b.x = w0[16 + m]; b.y = w1[16 + m];
    c1 = __builtin_amdgcn_wmma_f32_16x16x4_f32(false, a, false, b, (short)0, c1, false, false);
    b.x = w0[32 + m]; b.y = w1[32 + m];
    c2 = __builtin_amdgcn_wmma_f32_16x16x4_f32(false, a, false, b, (short)0, c2, false, false);
    b.x = w0[48 + m]; b.y = w1[48 + m];
    c3 = __builtin_amdgcn_wmma_f32_16x16x4_f32(false, a, false, b, (short)0, c3, false, false);
  }

  // C/D layout: VGPR r holds row (r + 8*kh), column m (within each 16-col tile)
#pragma unroll
  for (int r = 0; r < 8; ++r) {
    float* o = hout + (size_t)(rowBase + r + 8 * kh) * HC + m;
    o[0]  = c0[r];
    o[16] = c1[r];
    o[32] = c2[r];
    o[48] = c3[r];
  }
}

// ---------- per-node attention dot products ----------
__global__ void egat_attdots(const float* __restrict__ hbuf,
                             const float* __restrict__ a_dst_l,  // [H,C]
                             const float* __restrict__ a_src_l,  // [H,C]
                             float* __restrict__ s_dst, float* __restrict__ s_src) {
  int t = blockIdx.x * blockDim.x + threadIdx.x;
  if (t >= NN * HH) return;
  int n = t >> 1, h = t & 1;
  const float* hv = hbuf + (size_t)n * HC + h * CC;
  const float* ad = a_dst_l + h * CC;
  const float* as = a_src_l + h * CC;
  float sd = 0.f, ss = 0.f;
#pragma unroll
  for (int c = 0; c < CC; ++c) { sd += hv[c] * ad[c]; ss += hv[c] * as[c]; }
  s_dst[t] = sd;
  s_src[t] = ss;
}

// ---------- per-edge-class attention contribution ----------
__global__ void egat_esk(const float* __restrict__ eemb_l,   // [ECLS,64]
                         const float* __restrict__ a_src_l,  // [H,C]
                         float* __restrict__ e_s) {           // [ECLS,H]
  int t = threadIdx.x;
  if (t >= ECLS * HH) return;
  int cls = t >> 1, h = t & 1;
  const float* ev = eemb_l + cls * HC + h * CC;
  const float* as = a_src_l + h * CC;
  float s = 0.f;
#pragma unroll
  for (int c = 0; c < CC; ++c) s += ev[c] * as[c];
  e_s[t] = s;
}

// ---------- init per-layer accumulators ----------
__global__ void egat_init(float* __restrict__ agg, unsigned* __restrict__ amax,
                          float* __restrict__ denom) {
  int t = blockIdx.x * blockDim.x + threadIdx.x;
  if (t >= NN * HC) return;
  agg[t] = 0.f;
  if (t < NN * HH) { amax[t] = ENC_NEG_INF; denom[t] = 0.f; }
}

// ---------- edge pass 1: logits + segment max ----------
__global__ void egat_edge1(const int* __restrict__ srcArr, const int* __restrict__ dstArr,
                           const int* __restrict__ attr,
                           const float* __restrict__ s_dst, const float* __restrict__ s_src,
                           const float* __restrict__ e_s,
                           float* __restrict__ lgbuf, unsigned* __restrict__ amax) {
  int t = blockIdx.x * blockDim.x + threadIdx.x;
  if (t >= EE * HH) return;
  int e = t >> 1, h = t & 1;
  int s = srcArr[e], d = dstArr[e], a = attr[e];
  float lg = s_dst[d * HH + h] + s_src[s * HH + h] + e_s[a * HH + h];
  lg = (lg > 0.f) ? lg : 0.2f * lg;          // leaky_relu(0.2)
  lgbuf[t] = lg;
  atomicMax(&amax[d * HH + h], enc_ord(lg));
}

// ---------- fix segment max (replace -inf with 0), decode in place ----------
__global__ void egat_fixmax(unsigned* __restrict__ amax) {
  int t = blockIdx.x * blockDim.x + threadIdx.x;
  if (t >= NN * HH) return;
  unsigned u = amax[t];
  float m = (u == ENC_NEG_INF) ? 0.f : dec_ord(u);
  ((float*)amax)[t] = m;
}

// ---------- edge pass 2: exp + segment sum ----------
__global__ void egat_edge2(const int* __restrict__ dstArr,
                           const float* __restrict__ amaxf,
                           float* __restrict__ lgbuf, float* __restrict__ denom) {
  int t = blockIdx.x * blockDim.x + threadIdx.x;
  if (t >= EE * HH) return;
  int e = t >> 1, h = t & 1;
  int d = dstArr[e];
  float ex = expf(lgbuf[t] - amaxf[d * HH + h]);
  lgbuf[t] = ex;
  atomicAdd(&denom[d * HH + h], ex);
}

// ---------- edge pass 3: weighted aggregation ----------
__global__ void egat_edge3(const int* __restrict__ srcArr, const int* __restrict__ dstArr,
                           const int* __restrict__ attr,
                           const float* __restrict__ exbuf, const float* __restrict__ denom,
                           const float* __restrict__ hbuf, const float* __restrict__ eemb_l,
                           float* __restrict__ agg) {
  int t = blockIdx.x * blockDim.x + threadIdx.x;
  if (t >= EE * HC) return;
  int e = t >> 6, c = t & 63;
  int h = c >> 5;
  int s = srcArr[e], d = dstArr[e], a = attr[e];
  float alpha = exbuf[e * HH + h] / (denom[d * HH + h] + 1e-16f);
  float val = alpha * (hbuf[(size_t)s * HC + c] + eemb_l[a * HC + c]);
  atomicAdd(&agg[(size_t)d * HC + c], val);
}

// ---------- finalize: bias + ELU ----------
__global__ void egat_final(const float* __restrict__ agg, const float* __restrict__ bias_l,
                           float* __restrict__ act) {
  int t = blockIdx.x * blockDim.x + threadIdx.x;
  if (t >= NN * HC) return;
  float v = agg[t] + bias_l[t & 63];
  act[t] = (v > 0.f) ? v : expm1f(v);         // elu
}

// ---------- MF decoder: dot(user, item) ----------
__global__ void egat_mfdot(const float* __restrict__ act, float* __restrict__ out) {
  int p = blockIdx.x * blockDim.x + threadIdx.x;
  if (p >= NN / 4) return;
  const float* u = act + (size_t)(4 * p + 0) * HC;
  const float* v = act + (size_t)(4 * p + 1) * HC;
  float s = 0.f;
#pragma unroll
  for (int c = 0; c < HC; ++c) s += u[c] * v[c];
  out[p] = s;
}

extern "C" void kernel_launch(void* const* d_in, const int* in_sizes, int n_in,
                              void* d_out, int out_size, void* d_ws, size_t ws_size,
                              hipStream_t stream) {
  (void)in_sizes; (void)n_in; (void)out_size; (void)ws_size;
  const float* x       = (const float*)d_in[0];   // [N,4]
  const float* W0      = (const float*)d_in[1];   // [4,64]
  const float* W13     = (const float*)d_in[2];   // [3,64,64]
  const float* eemb    = (const float*)d_in[3];   // [L,5,64]
  const float* att_src = (const float*)d_in[4];   // [L,2,32]
  const float* att_dst = (const float*)d_in[5];   // [L,2,32]
  const float* bias    = (const float*)d_in[6];   // [L,64]
  const int*   eidx    = (const int*)d_in[7];     // [2,E]
  const int*   eattr   = (const int*)d_in[8];     // [E]
  const int* srcArr = eidx;
  const int* dstArr = eidx + EE;

  // workspace carve-up (floats)
  float* act   = (float*)d_ws;                 // N*64
  float* hbuf  = act  + (size_t)NN * HC;       // N*64
  float* agg   = hbuf + (size_t)NN * HC;       // N*64
  float* s_dst = agg  + (size_t)NN * HC;       // N*2
  float* s_src = s_dst + (size_t)NN * HH;      // N*2
  unsigned* amax = (unsigned*)(s_src + (size_t)NN * HH);   // N*2
  float* denom = (float*)amax + (size_t)NN * HH;           // N*2
  float* lgbuf = denom + (size_t)NN * HH;      // E*2 (logits, then exp)
  float* e_s   = lgbuf + (size_t)EE * HH;      // 10

  const int B = 256;
  const int gN64 = (NN * HC + B - 1) / B;
  const int gNH  = (NN * HH + B - 1) / B;
  const int gEH  = (EE * HH + B - 1) / B;
  const int gE64 = (EE * HC + B - 1) / B;

  for (int l = 0; l < LL; ++l) {
    const float* eemb_l = eemb + (size_t)l * ECLS * HC;
    const float* as_l   = att_src + (size_t)l * HC;
    const float* ad_l   = att_dst + (size_t)l * HC;
    const float* b_l    = bias + (size_t)l * HC;

    // 1) projection h = act @ W  (layer 0 reads x directly)
    if (l == 0) {
      egat_proj0<<<gN64, B, 0, stream>>>(x, W0, hbuf);
    } else {
      const float* Wl = W13 + (size_t)(l - 1) * HC * HC;
      egat_proj_wmma<<<NN / 16, 32, 0, stream>>>(act, Wl, hbuf);  // 6250 tiles exactly
    }

    // 2) per-node and per-class attention precomputations
    egat_attdots<<<gNH, B, 0, stream>>>(hbuf, ad_l, as_l, s_dst, s_src);
    egat_esk<<<1, 16, 0, stream>>>(eemb_l, as_l, e_s);

    // 3) init accumulators
    egat_init<<<gN64, B, 0, stream>>>(agg, amax, denom);

    // 4) attention softmax + aggregation
    egat_edge1<<<gEH, B, 0, stream>>>(srcArr, dstArr, eattr, s_dst, s_src, e_s, lgbuf, amax);
    egat_fixmax<<<gNH, B, 0, stream>>>(amax);
    egat_edge2<<<gEH, B, 0, stream>>>(dstArr, (const float*)amax, lgbuf, denom);
    egat_edge3<<<gE64, B, 0, stream>>>(srcArr, dstArr, eattr, lgbuf, denom, hbuf, eemb_l, agg);

    // 5) bias + ELU -> next activations
    egat_final<<<gN64, B, 0, stream>>>(agg, b_l, act);
  }

  // decoder
  egat_mfdot<<<(NN / 4 + B - 1) / B, B, 0, stream>>>(act, (float*)d_out);
}